// MedianPooling_38491496907348
// MI455X (gfx1250) — compile-verified
//
#include <hip/hip_runtime.h>
#include <stdint.h>

// Problem geometry (fixed by the reference): x is (8,16,512,512) f32.
#define IMG_H 512
#define IMG_W 512
#define NPLANES (8 * 16)

// Tile geometry: 512 threads/block (16 wave32), tile = 256 wide x 16 tall.
#define TW 256
#define TH 16
#define TILES_X (IMG_W / TW)                 // 2
#define TILES_Y (IMG_H / TH)                 // 32
#define TILES_PER_PLANE (TILES_X * TILES_Y)  // 64

// LDS row layout (floats): [pad pad pad | Lhalo@3 | interior@4..259 | Rhalo@260 | pad pad pad]
// Row stride 264 floats = 1056 B = 66*16 B -> interior (offset +16 B) is
// 16-B aligned on EVERY row, enabling b128 async loads for the interior.
#define LSTRIDE 264
#define LH (TH + 2)                // 18 halo rows
#define BLOCK 512
#define EDGE_TASKS (LH * 2)        // 36 b32 tasks (left/right halo columns)

__device__ __forceinline__ int reflect_idx(int i, int n) {
    i = (i < 0) ? -i : i;                 // reflect: -1 -> 1
    i = (i >= n) ? (2 * n - 2 - i) : i;   // reflect: n -> n-2
    return i;
}

__device__ __forceinline__ void sort2(float& a, float& b) {
    float lo = fminf(a, b);
    float hi = fmaxf(a, b);
    a = lo;
    b = hi;
}

__device__ __forceinline__ float med3(float a, float b, float c) {
    return fmaxf(fminf(a, b), fminf(fmaxf(a, b), c));
}

// One async 16-byte chunk: halo row lr, chunk q (column byte offset qb=16*q).
__device__ __forceinline__ void async_row_chunk(const float* src, float* ldsrow0,
                                                int gy0, int gx0, int lr, int qb4) {
    int gr = reflect_idx(gy0 + lr - 1, IMG_H);
    const float* gp = src + gr * IMG_W + gx0 + qb4;
    unsigned lds_addr = (unsigned)(size_t)(ldsrow0 + lr * LSTRIDE + 4 + qb4);
    asm volatile("global_load_async_to_lds_b128 %0, %1, off"
                 :: "v"(lds_addr), "v"(gp)
                 : "memory");
}

__global__ __launch_bounds__(BLOCK)
void median3x3_reflect_kernel(const float* __restrict__ x,
                              float* __restrict__ out) {
    __shared__ __align__(16) float tile[LH * LSTRIDE];

    const int t = threadIdx.x;
    const int bid = blockIdx.x;
    // All-constant tile math: shifts and masks only.
    const int plane = bid >> 6;            // / TILES_PER_PLANE (64)
    const int rem   = bid & 63;
    const int ty    = rem >> 1;            // / TILES_X (2)
    const int tx    = rem & 1;
    const int gx0   = tx * TW;
    const int gy0   = ty * TH;

    const float* __restrict__ src = x + (size_t)plane * (IMG_H * IMG_W);
    float* __restrict__ dst = out + (size_t)plane * (IMG_H * IMG_W);

    // ---------------- Phase 1a: interior, async b128 global -> LDS ----------
    // 18 rows x 64 chunks = 1152 tasks over 512 threads: each thread owns
    // chunk q = t&63 on rows t>>6, 8+(t>>6), and (t<128 only) 16+(t>>6).
    // Straight-line: three async issues, no loop/exec-mask overhead. Every
    // wave's 32 lanes cover 512 contiguous bytes of one row -> fully
    // coalesced; reflect remap only changes which ROW is fetched.
    {
        const int q   = t & 63;
        const int qb4 = q << 2;            // column offset in floats
        const int r   = t >> 6;            // 0..7
        async_row_chunk(src, tile, gy0, gx0, r, qb4);
        async_row_chunk(src, tile, gy0, gx0, r + 8, qb4);
        if (t < 128)                       // rows 16,17
            async_row_chunk(src, tile, gy0, gx0, r + 16, qb4);
    }

    // ---------------- Phase 1b: halo columns, async b32 ---------------------
    if (t < EDGE_TASKS) {
        int lr   = t >> 1;
        int side = t & 1;                  // 0 = left halo, 1 = right halo
        int gr = reflect_idx(gy0 + lr - 1, IMG_H);
        int gc = reflect_idx(gx0 + (side ? TW : -1), IMG_W);
        const float* gp = src + gr * IMG_W + gc;
        unsigned lds_addr =
            (unsigned)(size_t)(&tile[lr * LSTRIDE + (side ? (4 + TW) : 3)]);
        asm volatile("global_load_async_to_lds_b32 %0, %1, off"
                     :: "v"(lds_addr), "v"(gp)
                     : "memory");
    }

    // Wait for this wave's async transfers, then sync the workgroup so every
    // wave sees the full LDS tile.
    asm volatile("s_wait_asynccnt 0" ::: "memory");
    __syncthreads();

    // ---------------- Phase 2: median-of-9 from LDS -------------------------
    // Thread t owns output column c of the tile and rows [r0, r0+8).
    const int c  = t & (TW - 1);   // 0..255
    const int rg = t >> 8;         // 0 or 1
    const int r0 = rg * 8;

    // Read the 10x3 register window (window for col c = LDS cols 3+c..5+c,
    // contiguous, conflict-free) and sort each horizontal triple ONCE; each
    // sorted triple is shared by up to 3 of this thread's output rows.
    float lo[10], mi[10], hi[10];
#pragma unroll
    for (int r = 0; r < 10; ++r) {
        const float* p = &tile[(r0 + r) * LSTRIDE + 3 + c];
        float a = p[0];
        float b = p[1];
        float d = p[2];
        sort2(a, b);
        sort2(b, d);
        sort2(a, b);
        lo[r] = a;
        mi[r] = b;
        hi[r] = d;
    }

    // median9 = med3( max(lo0,lo1,lo2), med3(mi0,mi1,mi2), min(hi0,hi1,hi2) )
#pragma unroll
    for (int r = 0; r < 8; ++r) {
        float mn = fmaxf(fmaxf(lo[r], lo[r + 1]), lo[r + 2]);
        float md = med3(mi[r], mi[r + 1], mi[r + 2]);
        float mx = fminf(fminf(hi[r], hi[r + 1]), hi[r + 2]);
        float m  = med3(mn, md, mx);
        // Write-once stream: non-temporal so it doesn't evict the input in L2.
        __builtin_nontemporal_store(
            m, &dst[(size_t)(gy0 + r0 + r) * IMG_W + (gx0 + c)]);
    }
}

extern "C" void kernel_launch(void* const* d_in, const int* in_sizes, int n_in,
                              void* d_out, int out_size, void* d_ws, size_t ws_size,
                              hipStream_t stream) {
    const float* x = (const float*)d_in[0];
    float* out = (float*)d_out;

    const int nblocks = NPLANES * TILES_PER_PLANE;  // 8192
    median3x3_reflect_kernel<<<nblocks, BLOCK, 0, stream>>>(x, out);
}